// GRULayer_58411555225821
// MI455X (gfx1250) — compile-verified
//
#include <hip/hip_runtime.h>
#include <math.h>

// GRU dims (fixed by the reference)
#define BB 64
#define TT 1024
#define DD 512
#define HH 1024
#define GG 3072   // 3*H
#define KC 32     // K-chunk staged in LDS
#define ROWP 36   // LDS row pitch (floats): 144B -> 16B-aligned rows,
                  // and 36*4B stride is bank-conflict-free across 16 lanes

typedef __attribute__((ext_vector_type(2))) float v2f;
typedef __attribute__((ext_vector_type(8))) float v8f;

// Async DMA: 16 bytes global -> LDS per lane (GLOBAL_LOAD_ASYNC_TO_LDS_B128).
// LDS operand is the flat pointer's low 32 bits (== wave-relative LDS addr).
__device__ __forceinline__ void async_copy16(const float* g, const float* l) {
    unsigned lds_addr = (unsigned)(size_t)l;
    asm volatile("global_load_async_to_lds_b128 %0, %1, off"
                 :: "v"(lds_addr), "v"(g) : "memory");
}

__device__ __forceinline__ void wait_async0() {
#if defined(__has_builtin) && __has_builtin(__builtin_amdgcn_s_wait_asynccnt)
    __builtin_amdgcn_s_wait_asynccnt(0);
#else
    asm volatile("s_wait_asynccnt 0x0" ::: "memory");
#endif
}

// Out[m, n] = sum_k A[m,k] * W[n,k] + bias[n]
//   A: M rows (stride lda floats) x K cols. A==nullptr -> zeros (bias only).
//   W: row-major [3072 x K]  (multiply by W^T).
// Block = 256 threads (8 waves). Block tile: 64 M x 128 N.
// Each wave: 16 N-columns x full 64 M -> 4 independent accumulators, so one
// B fragment feeds 4 back-to-back V_WMMA_F32_16X16X4_F32.
// Staging: double-buffered LDS filled by async global->LDS DMA (ASYNCcnt),
// one barrier per K-chunk, DMA of chunk c+1 overlaps WMMA of chunk c.
// Grid: x = N/128 (=24), y = M/64.
__global__ __launch_bounds__(256) void gemm_xwT_wmma(
    const float* __restrict__ A, long lda,
    const float* __restrict__ W, int K,
    const float* __restrict__ bias,
    float* __restrict__ Out, long ldo)
{
    __shared__ __align__(16) float As[2][64][ROWP];
    __shared__ __align__(16) float Ws[2][128][ROWP];

    const int tid  = threadIdx.x;
    const int lane = tid & 31;
    const int wave = tid >> 5;        // 0..7 -> 16-wide N subtile
    const int half = lane >> 4;       // K/M split per ISA VGPR layout
    const int l16  = lane & 15;

    const long m0 = (long)blockIdx.y * 64;
    const long n0 = (long)blockIdx.x * 128;

    v8f acc[4];
    #pragma unroll
    for (int mt = 0; mt < 4; ++mt)
        acc[mt] = (v8f){0.f, 0.f, 0.f, 0.f, 0.f, 0.f, 0.f, 0.f};

    // Cooperative DMA coordinates: A tile 64x32 (2 x b128/thread),
    // W tile 128x32 (4 x b128/thread)
    const int a_row = tid >> 2;           // 0..63
    const int a_col = (tid & 3) * 8;      // 0,8,16,24
    const int w_row = tid >> 1;           // 0..127
    const int w_col = (tid & 1) * 16;     // 0,16

    if (A != nullptr) {
        const int nch = K / KC;
        const float* __restrict__ arow = A + (m0 + a_row) * lda + a_col;
        const float* __restrict__ wrow = W + (n0 + w_row) * (long)K + w_col;

        // Prologue: DMA chunk 0 into buffer 0
        async_copy16(arow,     &As[0][a_row][a_col]);
        async_copy16(arow + 4, &As[0][a_row][a_col + 4]);
        async_copy16(wrow,      &Ws[0][w_row][w_col]);
        async_copy16(wrow + 4,  &Ws[0][w_row][w_col + 4]);
        async_copy16(wrow + 8,  &Ws[0][w_row][w_col + 8]);
        async_copy16(wrow + 12, &Ws[0][w_row][w_col + 12]);
        wait_async0();
        __syncthreads();

        for (int c = 0; c < nch; ++c) {
            const int cur = c & 1;
            const int nxt = cur ^ 1;

            // Kick DMA for next chunk (overlaps the WMMA stream below)
            if (c + 1 < nch) {
                const int kc = (c + 1) * KC;
                async_copy16(arow + kc,     &As[nxt][a_row][a_col]);
                async_copy16(arow + kc + 4, &As[nxt][a_row][a_col + 4]);
                async_copy16(wrow + kc,      &Ws[nxt][w_row][w_col]);
                async_copy16(wrow + kc + 4,  &Ws[nxt][w_row][w_col + 4]);
                async_copy16(wrow + kc + 8,  &Ws[nxt][w_row][w_col + 8]);
                async_copy16(wrow + kc + 12, &Ws[nxt][w_row][w_col + 12]);
            }

            // Hoisted per-chunk LDS bases: k-loop uses immediate offsets only
            const float* pB  = &Ws[cur][wave * 16 + l16][2 * half];
            const float* pA0 = &As[cur][ 0 + l16][2 * half];
            const float* pA1 = &As[cur][16 + l16][2 * half];
            const float* pA2 = &As[cur][32 + l16][2 * half];
            const float* pA3 = &As[cur][48 + l16][2 * half];

            // 8 k-steps x 4 WMMA; one B frag per step feeds 4 accumulators
            #pragma unroll
            for (int kk = 0; kk < KC; kk += 4) {
                v2f b, a0, a1, a2, a3;
                b.x  = pB[kk];  b.y  = pB[kk + 1];
                a0.x = pA0[kk]; a0.y = pA0[kk + 1];
                a1.x = pA1[kk]; a1.y = pA1[kk + 1];
                a2.x = pA2[kk]; a2.y = pA2[kk + 1];
                a3.x = pA3[kk]; a3.y = pA3[kk + 1];
                acc[0] = __builtin_amdgcn_wmma_f32_16x16x4_f32(
                    false, a0, false, b, (short)0, acc[0], false, false);
                acc[1] = __builtin_amdgcn_wmma_f32_16x16x4_f32(
                    false, a1, false, b, (short)0, acc[1], false, false);
                acc[2] = __builtin_amdgcn_wmma_f32_16x16x4_f32(
                    false, a2, false, b, (short)0, acc[2], false, false);
                acc[3] = __builtin_amdgcn_wmma_f32_16x16x4_f32(
                    false, a3, false, b, (short)0, acc[3], false, false);
            }

            // Own DMA landed; barrier publishes everyone's chunk c+1 and
            // guarantees all waves are done reading buffer `cur`.
            wait_async0();
            __syncthreads();
        }
    }

    // C/D layout: VGPR r -> M = half*8 + r ; N = n0 + wave*16 + l16
    const long ng = n0 + wave * 16 + l16;
    const float bv = bias[ng];
    #pragma unroll
    for (int mt = 0; mt < 4; ++mt) {
        #pragma unroll
        for (int r = 0; r < 8; ++r) {
            Out[(m0 + mt * 16 + half * 8 + r) * ldo + ng] = acc[mt][r] + bv;
        }
    }
}

// Fused gate update for one timestep.
__global__ __launch_bounds__(256) void gru_gates(
    const float* __restrict__ Gi, long gi_row_stride,
    const float* __restrict__ Gh,
    const float* __restrict__ hprev,
    float* __restrict__ out, int t, int write_final)
{
    const int idx = blockIdx.x * 256 + threadIdx.x;   // 0 .. B*H-1
    const int b = idx >> 10;         // H == 1024
    const int j = idx & (HH - 1);

    const float* gi = Gi + (long)b * gi_row_stride;
    const float* gh = Gh + (long)b * GG;

    const float i_r = gi[j];
    const float i_z = gi[HH + j];
    const float i_n = gi[2 * HH + j];
    const float h_r = gh[j];
    const float h_z = gh[HH + j];
    const float h_n = gh[2 * HH + j];
    const float hp  = hprev ? hprev[(long)b * (TT * HH) + j] : 0.0f;

    const float r = 1.0f / (1.0f + expf(-(i_r + h_r)));
    const float z = 1.0f / (1.0f + expf(-(i_z + h_z)));
    const float n = tanhf(i_n + r * h_n);
    const float hn = (1.0f - z) * n + z * hp;

    out[(long)b * (TT * HH) + (long)t * HH + j] = hn;
    if (write_final) {
        out[(long)BB * TT * HH + (long)b * HH + j] = hn;  // h_final
    }
}

extern "C" void kernel_launch(void* const* d_in, const int* in_sizes, int n_in,
                              void* d_out, int out_size, void* d_ws, size_t ws_size,
                              hipStream_t stream) {
    (void)in_sizes; (void)n_in; (void)out_size;
    const float* x   = (const float*)d_in[0];   // [B, T, D]
    const float* Wih = (const float*)d_in[1];   // [3H, D]
    const float* Whh = (const float*)d_in[2];   // [3H, H]
    const float* bih = (const float*)d_in[3];   // [3H]
    const float* bhh = (const float*)d_in[4];   // [3H]
    float* out = (float*)d_out;                 // [B,T,H] ++ [B,H]

    float* Gh = (float*)d_ws;                                    // [B x 3H]
    const size_t ghB     = (size_t)BB * GG * sizeof(float);      // ~786 KB
    const size_t giFullB = (size_t)BB * TT * GG * sizeof(float); // ~805 MB
    const bool full = ws_size >= ghB + giFullB;
    float* GiFull = (float*)((char*)d_ws + ghB);
    float* GiStep = GiFull;  // per-step [B x 3H] when workspace is small

    const dim3 blk(256);

    if (full) {
        // One big parallel GEMM for all input projections:
        // rows m = b*T + t of x (contiguous, lda = D)
        dim3 g(GG / 128, (BB * TT) / 64);  // 24 x 1024
        gemm_xwT_wmma<<<g, blk, 0, stream>>>(x, (long)DD, Wih, DD, bih,
                                             GiFull, (long)GG);
    }

    const dim3 gstep(GG / 128, BB / 64);        // 24 x 1
    const dim3 ggate((BB * HH) / 256);          // 256 blocks

    for (int t = 0; t < TT; ++t) {
        if (!full) {
            // gi_t = x_t @ W_ih^T + b_ih   (rows stride T*D across batch)
            gemm_xwT_wmma<<<gstep, blk, 0, stream>>>(
                x + (long)t * DD, (long)TT * DD, Wih, DD, bih, GiStep, (long)GG);
        }
        // gh_t = h_{t-1} @ W_hh^T + b_hh ; h_{t-1} read from out[:, t-1, :]
        const float* hp = (t == 0) ? nullptr : out + (long)(t - 1) * HH;
        gemm_xwT_wmma<<<gstep, blk, 0, stream>>>(
            hp, (long)TT * HH, Whh, HH, bhh, Gh, (long)GG);

        const float* gi   = full ? (GiFull + (long)t * GG) : GiStep;
        const long gi_rs  = full ? (long)TT * GG : (long)GG;
        gru_gates<<<ggate, blk, 0, stream>>>(gi, gi_rs, Gh, hp, out, t,
                                             (t == TT - 1) ? 1 : 0);
    }
}